// TransformerCell_51986284151081
// MI455X (gfx1250) — compile-verified
//
#include <hip/hip_runtime.h>
#include <math.h>
#include <stdint.h>

#define B_    32
#define T_    4096
#define D_    1024
#define H_    16
#define DH    64
#define KTOT  (T_ + 1)              // 4097 keys (previous + token)
#define NTILE ((KTOT + 15) / 16)    // 257 tiles of 16 keys
#define RSTRIDE 68                  // padded row stride in floats (16B aligned)
#define NWAVES 8
#define SCALE 0.125f                // 1/sqrt(64)

// Flip to 0 if the async-to-LDS mnemonic is rejected by the assembler.
#define USE_ASYNC_STAGE 1

typedef __attribute__((ext_vector_type(2))) float v2f;
typedef __attribute__((ext_vector_type(8))) float v8f;

#ifndef __has_builtin
#define __has_builtin(x) 0
#endif
#if defined(__HIP_DEVICE_COMPILE__) && __has_builtin(__builtin_amdgcn_wmma_f32_16x16x4_f32)
#define ATTN_USE_WMMA 1
#else
#define ATTN_USE_WMMA 0
#endif

// ---------------------------------------------------------------------------
// Kernel 1: keys = concat(previous, token) -> second output region of d_out.
// Pure streaming b128 copy; this is 2/3 of total HBM traffic (~1.07 GB).
// ---------------------------------------------------------------------------
__global__ __launch_bounds__(256)
void concat_copy_kernel(const float4* __restrict__ tok4,
                        const float4* __restrict__ prev4,
                        float4* __restrict__ keys4) {
  const int b = blockIdx.y;
  const unsigned per_batch = KTOT * (D_ / 4);     // 4097*256 = 1,048,832
  unsigned i = blockIdx.x * blockDim.x + threadIdx.x;
  const unsigned stride = gridDim.x * blockDim.x;
  const float4* pb = prev4 + (size_t)b * (T_ * (D_ / 4));
  const float4* tb = tok4 + (size_t)b * (D_ / 4);
  float4* kb = keys4 + (size_t)b * per_batch;
  for (; i < per_batch; i += stride) {
    unsigned t = i >> 8;                          // D/4 == 256
    float4 v = (t == (unsigned)T_) ? tb[i & 255u] : pb[i];
    kb[i] = v;
  }
}

// ---------------------------------------------------------------------------
// Async stage: 16 rows x 64 floats of K directly into LDS (no VGPR bounce).
// 8 x GLOBAL_LOAD_ASYNC_TO_LDS_B128 per wave; lanes 0-15 carry even rows,
// lanes 16-31 odd rows; each lane moves 16B. Tracked by ASYNCcnt.
// ---------------------------------------------------------------------------
#if USE_ASYNC_STAGE
__device__ __forceinline__ void stage_tile_async(const float* kbase, int k0,
                                                 uint32_t lds_base, int lane) {
  const int half = lane >> 4;
  const int c4 = (lane & 15) * 4;
#pragma unroll
  for (int rr = 0; rr < 8; ++rr) {
    int kg = k0 + 2 * rr + half;
    if (kg > KTOT - 1) kg = KTOT - 1;             // clamp addr; masked later
    const float* src = kbase + (size_t)kg * D_ + c4;
    uint32_t dst = lds_base + (uint32_t)(((2 * rr + half) * RSTRIDE + c4) * 4);
    asm volatile("global_load_async_to_lds_b128 %0, %1, off"
                 :: "v"(dst), "v"((unsigned long long)(uintptr_t)src)
                 : "memory");
  }
}
#endif

// ---------------------------------------------------------------------------
// Kernel 2: single-pass (online-softmax) attention. One workgroup per (b,h),
// 8 waves split the 257 key tiles; WMMA f32 16x16x4 accumulates out += p^T K.
// Double-buffered async-to-LDS pipeline: load tile j+1 while computing j.
// ---------------------------------------------------------------------------
__global__ __launch_bounds__(256)
void attn_kernel(const float* __restrict__ token,
                 const float* __restrict__ keys,   // B x KTOT x D
                 float* __restrict__ out) {        // B x D
  __shared__ float q_s[DH];
  __shared__ float ktile[NWAVES][2][16 * RSTRIDE];
  __shared__ float merge_out[NWAVES][DH];
  __shared__ float merge_m[NWAVES];
  __shared__ float merge_l[NWAVES];

  const int bh = blockIdx.x;
  const int b = bh >> 4;
  const int h = bh & 15;
  const int tid = threadIdx.x;
  const int wave = tid >> 5;
  const int lane = tid & 31;
  const int ky = lane & 15;        // key-in-tile this lane scores
  const int half = lane >> 4;      // 32-dim half / K-slot pair for WMMA

  if (tid < DH) q_s[tid] = token[b * D_ + h * DH + tid];
  __syncthreads();

  const float* kbase = keys + (size_t)b * KTOT * D_ + h * DH;

  float m = -1e30f, l = 0.f;
#if ATTN_USE_WMMA
  v8f acc0 = {}, acc1 = {}, acc2 = {}, acc3 = {};
#else
  float f0 = 0.f, f1 = 0.f;        // lane owns d = 2*lane, 2*lane+1
#endif

#if USE_ASYNC_STAGE
  const uint32_t lds_off0 = (uint32_t)(uintptr_t)&ktile[wave][0][0];
  const uint32_t lds_off1 = (uint32_t)(uintptr_t)&ktile[wave][1][0];
  stage_tile_async(kbase, wave * 16, lds_off0, lane);   // preload first tile
  int pb = 0;
#endif

  for (int tile = wave; tile < NTILE; tile += NWAVES) {
    const int k0 = tile * 16;
    const float* myk;

#if USE_ASYNC_STAGE
    {
      int nxt = tile + NWAVES;
      if (nxt > NTILE - 1) nxt = NTILE - 1;       // redundant tail reload, unused
      stage_tile_async(kbase, nxt * 16, pb ? lds_off0 : lds_off1, lane);
      // <=8 outstanding: the 8 loads for the *current* tile are complete
      // (async loads complete in order).
      asm volatile("s_wait_asynccnt 0x8" ::: "memory");
      myk = &ktile[wave][pb][0];
      pb ^= 1;
    }
#else
    {
      const int c4 = ky * 4;
#pragma unroll
      for (int rr = 0; rr < 8; ++rr) {
        int kg = k0 + 2 * rr + half;
        if (kg > KTOT - 1) kg = KTOT - 1;
        const float4* src = (const float4*)(kbase + (size_t)kg * D_ + c4);
        *(float4*)&ktile[wave][0][(2 * rr + half) * RSTRIDE + c4] = *src;
      }
      myk = &ktile[wave][0][0];
    }
#endif

    // Prefetch 2 pipeline stages ahead into cache (global_prefetch_b8).
    {
      int kpn = k0 + 32 * NWAVES + ky;
      if (kpn > KTOT - 1) kpn = KTOT - 1;
      __builtin_prefetch(kbase + (size_t)kpn * D_ + half * 32, 0, 1);
    }

    // Scores: lane computes half the dot for key `ky`, then combines halves.
    float s = 0.f;
    {
      const float4* krow = (const float4*)&myk[ky * RSTRIDE + half * 32];
      const float4* qrow = (const float4*)&q_s[half * 32];
#pragma unroll
      for (int c = 0; c < 8; ++c) {
        float4 kv = krow[c], qv = qrow[c];
        s = fmaf(kv.x, qv.x, s); s = fmaf(kv.y, qv.y, s);
        s = fmaf(kv.z, qv.z, s); s = fmaf(kv.w, qv.w, s);
      }
    }
    s += __shfl_xor(s, 16, 32);
    s *= SCALE;
    if (k0 + ky >= KTOT) s = -1e30f;              // mask tail keys

    // Online softmax. Max-update is wave-uniform -> skip rescale when stale.
    float mt = s;
#pragma unroll
    for (int off = 8; off >= 1; off >>= 1) mt = fmaxf(mt, __shfl_xor(mt, off, 32));
    const float m_new = fmaxf(m, mt);
    if (m_new > m) {                              // uniform branch
      const float corr = __expf(m - m_new);
      l *= corr;
#if ATTN_USE_WMMA
      acc0 = acc0 * corr; acc1 = acc1 * corr;
      acc2 = acc2 * corr; acc3 = acc3 * corr;
#else
      f0 *= corr; f1 *= corr;
#endif
      m = m_new;
    }
    const float p = __expf(s - m);                // masked keys -> exactly 0
    float ps = p;
#pragma unroll
    for (int off = 8; off >= 1; off >>= 1) ps += __shfl_xor(ps, off, 32);
    l += ps;

#if ATTN_USE_WMMA
    // out(1x64) += p(1x16) * K_tile(16x64) via chained V_WMMA_F32_16X16X4_F32.
    // A rows identical (broadcast p chunk); each C covers 16 d-columns.
#pragma unroll
    for (int g = 0; g < 4; ++g) {
      const int srcl = g * 4 + half * 2;          // K-slot mapping per ISA layout
      v2f a;
      a.x = __shfl(p, srcl, 32);
      a.y = __shfl(p, srcl + 1, 32);
      const float* r0p = &myk[(g * 4 + half * 2) * RSTRIDE + ky];
      const float* r1p = r0p + RSTRIDE;
      v2f bb;
      bb.x = r0p[0];  bb.y = r1p[0];
      acc0 = __builtin_amdgcn_wmma_f32_16x16x4_f32(false, a, false, bb, (short)0, acc0, false, false);
      bb.x = r0p[16]; bb.y = r1p[16];
      acc1 = __builtin_amdgcn_wmma_f32_16x16x4_f32(false, a, false, bb, (short)0, acc1, false, false);
      bb.x = r0p[32]; bb.y = r1p[32];
      acc2 = __builtin_amdgcn_wmma_f32_16x16x4_f32(false, a, false, bb, (short)0, acc2, false, false);
      bb.x = r0p[48]; bb.y = r1p[48];
      acc3 = __builtin_amdgcn_wmma_f32_16x16x4_f32(false, a, false, bb, (short)0, acc3, false, false);
    }
#else
#pragma unroll
    for (int k = 0; k < 16; ++k) {
      float pk = __shfl(p, k, 32);
      f0 = fmaf(pk, myk[k * RSTRIDE + 2 * lane + 0], f0);
      f1 = fmaf(pk, myk[k * RSTRIDE + 2 * lane + 1], f1);
    }
#endif
  }

#if USE_ASYNC_STAGE
  asm volatile("s_wait_asynccnt 0x0" ::: "memory");
#endif

  // Per-wave partials -> LDS.
  if (lane == 0) { merge_m[wave] = m; merge_l[wave] = l; }
#if ATTN_USE_WMMA
  if (lane < 16) {                                // all C rows identical; comp 0
    merge_out[wave][ 0 + lane] = acc0[0];
    merge_out[wave][16 + lane] = acc1[0];
    merge_out[wave][32 + lane] = acc2[0];
    merge_out[wave][48 + lane] = acc3[0];
  }
#else
  merge_out[wave][2 * lane + 0] = f0;
  merge_out[wave][2 * lane + 1] = f1;
#endif
  __syncthreads();

  // Log-sum-exp merge of the 8 wave partials; 64 threads, one per d.
  if (tid < DH) {
    float M = merge_m[0];
#pragma unroll
    for (int w = 1; w < NWAVES; ++w) M = fmaxf(M, merge_m[w]);
    float L = 0.f, val = 0.f;
#pragma unroll
    for (int w = 0; w < NWAVES; ++w) {
      float e = __expf(merge_m[w] - M);
      L += merge_l[w] * e;
      val = fmaf(merge_out[w][tid], e, val);
    }
    out[b * D_ + h * DH + tid] = val / L;
  }
}

// ---------------------------------------------------------------------------
extern "C" void kernel_launch(void* const* d_in, const int* in_sizes, int n_in,
                              void* d_out, int out_size, void* d_ws, size_t ws_size,
                              hipStream_t stream) {
  (void)in_sizes; (void)n_in; (void)d_ws; (void)ws_size; (void)out_size;
  const float* token    = (const float*)d_in[0];
  const float* previous = (const float*)d_in[1];
  float* out  = (float*)d_out;                    // result: B*1*D = 32768 floats
  float* keys = out + (size_t)B_ * D_;            // next_previous: B*KTOT*D

  dim3 cgrid(1024, B_);
  concat_copy_kernel<<<cgrid, 256, 0, stream>>>((const float4*)token,
                                                (const float4*)previous,
                                                (float4*)keys);
  attn_kernel<<<dim3(B_ * H_), dim3(256), 0, stream>>>(token, keys, out);
}